// RSNN_43018392437486
// MI455X (gfx1250) — compile-verified
//
#include <hip/hip_runtime.h>

typedef __attribute__((ext_vector_type(2))) float v2f;
typedef __attribute__((ext_vector_type(8))) float v8f;

#define THRESH 0.3f
#define BETA   0.1f
#define ALPHA  0.2997469f   /* expf(-1.0f/0.83f) */
#define TWIN   50
#define HID    1024
#define NOUT   10
#define BT     16           /* batch rows per recurrent workgroup */

/* ------------------------------------------------------------------ */
/* CDNA5 async global->LDS copy (ASYNCcnt path).                       */
/* LDS address is taken from the VDST VGPR; low 32 bits of a flat LDS      */
/* pointer are the wave-relative LDS offset, so ptr->u32 is the VDST.  */
/* ------------------------------------------------------------------ */
__device__ __forceinline__ void async_copy_b128(const void* gsrc, void* ldst) {
  unsigned loff = (unsigned)(uintptr_t)ldst;
  asm volatile("global_load_async_to_lds_b128 %0, %1, off"
               :: "v"(loff), "v"(gsrc) : "memory");
}
__device__ __forceinline__ void wait_async0() {
  asm volatile("s_wait_asynccnt 0" ::: "memory");
}

/* ------------------------------------------------------------------ */
/* Kernel 1: drive[b*T+t, h] = sum_i x[b,t,i] * W_ih[h,i]              */
/* M = B*T (51200), N = H (1024), K = I (2312)                         */
/* block tile 64(M) x 256(N); 8 waves of 32x64 (8 wmma per 6 DS loads) */
/* K chunks of 16, double-buffered LDS fed by async-to-LDS DMA.        */
/* ------------------------------------------------------------------ */
#define TM 64
#define TN 256
#define TK 16
#define PAD 4      /* row stride 20 floats = 80B: 16B aligned, conflict-free */

__global__ __launch_bounds__(256) void drive_gemm(
    const float* __restrict__ X, const float* __restrict__ Wih,
    float* __restrict__ drive, int M, int N, int K) {
  __shared__ __align__(16) float As[2][TM][TK + PAD];
  __shared__ __align__(16) float Bs[2][TN][TK + PAD];

  const int tid  = threadIdx.x;
  const int wid  = tid >> 5;
  const int lane = tid & 31;
  const int lr   = lane & 15;
  const int hi   = lane >> 4;
  const int bm   = blockIdx.y * TM;
  const int bn   = blockIdx.x * TN;
  const int wm   = (wid & 1) * 32;   /* wave M origin: 2 waves in M */
  const int wn   = (wid >> 1) * 64;  /* wave N origin: 4 waves in N */

  const v8f zero8 = {0.f, 0.f, 0.f, 0.f, 0.f, 0.f, 0.f, 0.f};
  v8f acc[2][4];
#pragma unroll
  for (int i = 0; i < 2; ++i)
#pragma unroll
    for (int j = 0; j < 4; ++j) acc[i][j] = zero8;

  /* per-thread async slots: A = 1 x b128, B = 4 x b128 per chunk */
  const int arow = tid >> 2, ac4 = (tid & 3) << 2;

  const int Kmain = (K / TK) * TK;   /* 2304 */
  const int NC    = Kmain / TK;      /* 144  */

  /* prologue: stage chunk 0 */
  {
    async_copy_b128(X + (size_t)(bm + arow) * K + ac4, &As[0][arow][ac4]);
#pragma unroll
    for (int i = 0; i < 4; ++i) {
      int s = tid + (i << 8);
      int row = s >> 2, c4 = (s & 3) << 2;
      async_copy_b128(Wih + (size_t)(bn + row) * K + c4, &Bs[0][row][c4]);
    }
  }

  for (int c = 0; c < NC; ++c) {
    wait_async0();       /* chunk c fully landed in LDS (in-order done) */
    __syncthreads();     /* everyone done reading buf[(c+1)&1] (iter c-1) */

    if (c + 1 < NC) {    /* DMA chunk c+1 overlaps compute of chunk c */
      int k0 = (c + 1) * TK;
      async_copy_b128(X + (size_t)(bm + arow) * K + k0 + ac4,
                      &As[(c + 1) & 1][arow][ac4]);
#pragma unroll
      for (int i = 0; i < 4; ++i) {
        int s = tid + (i << 8);
        int row = s >> 2, c4 = (s & 3) << 2;
        async_copy_b128(Wih + (size_t)(bn + row) * K + k0 + c4,
                        &Bs[(c + 1) & 1][row][c4]);
      }
    }

    const float (*A)[TK + PAD]  = As[c & 1];
    const float (*Bt)[TK + PAD] = Bs[c & 1];
#pragma unroll
    for (int kk = 0; kk < TK; kk += 4) {
      v2f a[2], b[4];
#pragma unroll
      for (int i = 0; i < 2; ++i) {
        a[i].x = A[wm + i * 16 + lr][kk + 2 * hi];
        a[i].y = A[wm + i * 16 + lr][kk + 2 * hi + 1];
      }
#pragma unroll
      for (int j = 0; j < 4; ++j) {
        b[j].x = Bt[wn + j * 16 + lr][kk + 2 * hi];
        b[j].y = Bt[wn + j * 16 + lr][kk + 2 * hi + 1];
      }
#pragma unroll
      for (int i = 0; i < 2; ++i)
#pragma unroll
        for (int j = 0; j < 4; ++j)
          acc[i][j] = __builtin_amdgcn_wmma_f32_16x16x4_f32(
              false, a[i], false, b[j], (short)0, acc[i][j], false, false);
    }
  }

  /* ---- tail: K remainder (8), zero-padded synchronous staging ---- */
  __syncthreads();
  {
#pragma unroll
    for (int e = 0; e < 4; ++e) {
      int gk = Kmain + ac4 + e;
      As[0][arow][ac4 + e] = (gk < K) ? X[(size_t)(bm + arow) * K + gk] : 0.f;
    }
#pragma unroll
    for (int i = 0; i < 4; ++i) {
      int s = tid + (i << 8);
      int row = s >> 2, c4 = (s & 3) << 2;
#pragma unroll
      for (int e = 0; e < 4; ++e) {
        int gk = Kmain + c4 + e;
        Bs[0][row][c4 + e] = (gk < K) ? Wih[(size_t)(bn + row) * K + gk] : 0.f;
      }
    }
  }
  __syncthreads();
#pragma unroll
  for (int kk = 0; kk < TK; kk += 4) {
    v2f a[2], b[4];
#pragma unroll
    for (int i = 0; i < 2; ++i) {
      a[i].x = As[0][wm + i * 16 + lr][kk + 2 * hi];
      a[i].y = As[0][wm + i * 16 + lr][kk + 2 * hi + 1];
    }
#pragma unroll
    for (int j = 0; j < 4; ++j) {
      b[j].x = Bs[0][wn + j * 16 + lr][kk + 2 * hi];
      b[j].y = Bs[0][wn + j * 16 + lr][kk + 2 * hi + 1];
    }
#pragma unroll
    for (int i = 0; i < 2; ++i)
#pragma unroll
      for (int j = 0; j < 4; ++j)
        acc[i][j] = __builtin_amdgcn_wmma_f32_16x16x4_f32(
            false, a[i], false, b[j], (short)0, acc[i][j], false, false);
  }

#pragma unroll
  for (int i = 0; i < 2; ++i)
#pragma unroll
    for (int j = 0; j < 4; ++j)
#pragma unroll
      for (int r = 0; r < 8; ++r) {
        int row = bm + wm + i * 16 + 8 * hi + r;
        int col = bn + wn + j * 16 + lr;
        drive[(size_t)row * N + col] = acc[i][j][r];
      }
}

/* ------------------------------------------------------------------ */
/* Kernel 2: per-sample LIF recurrence. One WG owns BT=16 batch rows.  */
/* spike LDS is K-major [h][b] (64KB): conflict-free A-fragment reads. */
/* h_mem stays in the owning lanes' registers across all 50 steps.     */
/* ------------------------------------------------------------------ */
__global__ __launch_bounds__(256) void rsnn_recur(
    const float* __restrict__ drive,  /* [B*T, H], row = b*T + t */
    const float* __restrict__ Whh,    /* [H, H]  */
    const float* __restrict__ Who,    /* [O, H]  */
    float* __restrict__ out,          /* [B, O]  */
    int Tlen) {
  __shared__ float spike[HID][BT];    /* 64 KB */

  const int tid   = threadIdx.x;
  const int wid   = tid >> 5;
  const int lane  = tid & 31;
  const int lr    = lane & 15;
  const int hi    = lane >> 4;
  const int bBase = blockIdx.x * BT;

  for (int i = tid; i < HID * BT; i += 256) (&spike[0][0])[i] = 0.f;

  float hmem[8][8];
#pragma unroll
  for (int tt = 0; tt < 8; ++tt)
#pragma unroll
    for (int r = 0; r < 8; ++r) hmem[tt][r] = 0.f;

  float o_mem = 0.f, o_spk = 0.f, o_sum = 0.f;
  const int ob = tid / NOUT, oo = tid % NOUT; /* used when tid < 160 */

  const v8f zero8 = {0.f, 0.f, 0.f, 0.f, 0.f, 0.f, 0.f, 0.f};
  __syncthreads();

  for (int t = 0; t < Tlen; ++t) {
    /* --- Phase A: rec = spike(16xH) @ W_hh^T, wave owns 8 N-tiles --- */
    v8f acc[8];
#pragma unroll
    for (int tt = 0; tt < 8; ++tt) acc[tt] = zero8;

    for (int k = 0; k < HID; k += 4) {
      v2f a;
      a.x = spike[k + 2 * hi][lr];
      a.y = spike[k + 2 * hi + 1][lr];
#pragma unroll
      for (int tt = 0; tt < 8; ++tt) {
        int n = (wid * 8 + tt) * 16 + lr;
        const float* wp = Whh + (size_t)n * HID + k + 2 * hi;
        v2f b; b.x = wp[0]; b.y = wp[1];
        acc[tt] = __builtin_amdgcn_wmma_f32_16x16x4_f32(false, a, false, b,
                    (short)0, acc[tt], false, false);
      }
    }
    __syncthreads();  /* all spike reads for this step complete */

    /* --- Phase B: LIF update, write new spikes --- */
#pragma unroll
    for (int tt = 0; tt < 8; ++tt) {
      int n = (wid * 8 + tt) * 16 + lr;
#pragma unroll
      for (int r = 0; r < 8; ++r) {
        int m = 8 * hi + r;
        float drv = drive[(size_t)((bBase + m) * Tlen + t) * HID + n];
        float osp = spike[n][m];
        float pre = drv + acc[tt][r] + hmem[tt][r] * (ALPHA * (1.f - osp))
                    - BETA * osp;
        float s = (pre > THRESH) ? 1.f : 0.f;
        hmem[tt][r] = (pre < THRESH) ? pre : 0.f;
        spike[n][m] = s;
      }
    }
    __syncthreads();  /* new spikes visible */

    /* --- Phase C: output LIF (160 threads = 5 full waves, uniform) --- */
    if (tid < BT * NOUT) {
      float dot = 0.f;
      const float4* wo4 = (const float4*)(Who + oo * HID);
      for (int j4 = 0; j4 < HID / 4; ++j4) {
        float4 w = wo4[j4];
        int j = j4 * 4;
        dot += spike[j + 0][ob] * w.x + spike[j + 1][ob] * w.y
             + spike[j + 2][ob] * w.z + spike[j + 3][ob] * w.w;
      }
      o_mem = o_mem * (ALPHA * (1.f - o_spk)) + dot;
      o_spk = (o_mem > THRESH) ? 1.f : 0.f;
      o_mem = (o_mem < THRESH) ? o_mem : 0.f;
      o_sum += o_spk;
    }
    /* next write to spike happens only after the Phase-A barrier */
  }

  if (tid < BT * NOUT)
    out[(size_t)(bBase + ob) * NOUT + oo] = o_sum * (1.f / (float)TWIN);
}

/* ------------------------------------------------------------------ */
extern "C" void kernel_launch(void* const* d_in, const int* in_sizes, int n_in,
                              void* d_out, int out_size, void* d_ws, size_t ws_size,
                              hipStream_t stream) {
  const float* x   = (const float*)d_in[0];  /* [B, T, I] */
  const float* Wih = (const float*)d_in[1];  /* [H, I]    */
  const float* Whh = (const float*)d_in[2];  /* [H, H]    */
  const float* Who = (const float*)d_in[3];  /* [O, H]    */
  float* outp  = (float*)d_out;
  float* drive = (float*)d_ws;               /* [B*T, H] = 200 MB scratch */

  const int I = 2312;
  const int B = in_sizes[0] / (TWIN * I);    /* 1024 */
  const int M = B * TWIN;                    /* 51200 */

  dim3 g1(HID / TN, M / TM);                 /* 4 x 800 */
  drive_gemm<<<g1, 256, 0, stream>>>(x, Wih, drive, M, HID, I);

  rsnn_recur<<<dim3(B / BT), 256, 0, stream>>>(drive, Whh, Who, outp, TWIN);
}